// MultiHeadAttention_SMC_64622077936273
// MI455X (gfx1250) — compile-verified
//
#include <hip/hip_runtime.h>
#include <hip/hip_bf16.h>

// ---------------------------------------------------------------------------
// MultiHeadAttention_SMC  (B=8, P=16, S=1024, D=1024, H=16, dh=64, sigma=0.1)
// Memory-bound (~2.2 GB min traffic @ 23.3 TB/s ~= 95us). Strategy:
//  - fuse K_out copy with logit computation (read K exactly once)
//  - fuse V_out copy with zh accumulation (read V exactly once)
//  - fp32 WMMA (v_wmma_f32_16x16x4_f32) for the 4 projection GEMMs
//  - no pointer-selects anywhere so address-space inference gives
//    global_load/global_store (not flat_*) on all hot paths
// ---------------------------------------------------------------------------

#define BP 128          // B*P
#define Sc 1024
#define Dc 1024
#define Hc 16
#define SIGMA 0.1f
#define SCALE 0.125f    // 1/sqrt(64)
#define SCHUNK 128      // rows of S per streaming block
#define NCHUNK 8        // S / SCHUNK

typedef float v2f __attribute__((ext_vector_type(2)));
typedef float v8f __attribute__((ext_vector_type(8)));

// ---------------------------------------------------------------------------
// GEMM: O[m][n] = sum_k X[m][k] * W[k][n] + Bi[n] + SIGMA*Nz[m][n]
// M = 128, N = K = 1024.  One wave computes a 16x16 tile with
// V_WMMA_F32_16X16X4_F32, looping k in steps of 4 (256 WMMAs per tile).
// Launched once per matrix (q/k/v/z) so all pointers are direct kernel args.
// ---------------------------------------------------------------------------
__global__ __launch_bounds__(32)
void proj_gemm_wmma(const float* __restrict__ X, const float* __restrict__ W,
                    const float* __restrict__ Bi, const float* __restrict__ Nz,
                    float* __restrict__ O)
{
    const int lane = threadIdx.x;          // 0..31
    const int r    = lane & 15;            // row-in-tile (A) / col-in-tile (B,D)
    const int kg   = lane >> 4;            // K-pair group: 0 -> K{0,1}, 1 -> K{2,3}
    const int M0   = blockIdx.y * 16;
    const int N0   = blockIdx.x * 16;

    v8f acc = {0.f, 0.f, 0.f, 0.f, 0.f, 0.f, 0.f, 0.f};

    const size_t arow = (size_t)(M0 + r) * Dc;

#pragma unroll 4
    for (int k0 = 0; k0 < Dc; k0 += 4) {
        // A fragment: lane holds X[M0+r][k0+2*kg .. +1]  (8B aligned)
        v2f fa = *reinterpret_cast<const v2f*>(X + arow + k0 + 2 * kg);
        // B fragment: lane holds W[k0+2*kg][N0+r], W[k0+2*kg+1][N0+r]
        const size_t brow = (size_t)(k0 + 2 * kg) * Dc + N0 + r;
        v2f fb;
        fb.x = W[brow];
        fb.y = W[brow + Dc];
        acc = __builtin_amdgcn_wmma_f32_16x16x4_f32(
            false, fa, false, fb, (short)0, acc, false, false);
    }

    // D layout: VGPR j -> row M0+j (lanes 0-15) / M0+j+8 (lanes 16-31), col N0+lane%16
#pragma unroll
    for (int j = 0; j < 8; ++j) {
        const int row = M0 + j + kg * 8;
        const int col = N0 + r;
        const size_t idx = (size_t)row * Dc + col;
        O[idx] = acc[j] + Bi[col] + SIGMA * Nz[idx];
    }
}

// ---------------------------------------------------------------------------
// K pass: stream 128 rows of K per block; write K_out (row t replaced by kp
// via a DATA select, not a pointer select), and compute per-head scaled
// logits with 16-lane shuffle reductions (wave32: a head never crosses a wave).
// grid = (NCHUNK, BP), block = 256 threads (thread t -> 4 contiguous cols).
// ---------------------------------------------------------------------------
__global__ __launch_bounds__(256)
void attn_k_pass(const float* __restrict__ Kin, const float* __restrict__ kp,
                 const float* __restrict__ qp, const int* __restrict__ tptr,
                 float* __restrict__ Kout, float* __restrict__ logits)
{
    const int bp     = blockIdx.y;
    const int s0     = blockIdx.x * SCHUNK;
    const int tid    = threadIdx.x;         // 0..255
    const int col4   = tid * 4;             // 0..1020, within one head (64 cols)
    const int head   = tid >> 4;            // 16 threads per head
    const int lane16 = tid & 15;
    const int tstep  = *tptr;

    // q fragment for this thread's 4 columns, pre-scaled
    const float* qrow = qp + (size_t)bp * Dc + col4;
    const float q0 = qrow[0] * SCALE, q1 = qrow[1] * SCALE;
    const float q2 = qrow[2] * SCALE, q3 = qrow[3] * SCALE;

    const float* kprow = kp + (size_t)bp * Dc + col4;
    const size_t rowbase = (size_t)bp * Sc;

    for (int i = 0; i < SCHUNK; ++i) {
        const int s = s0 + i;
        if (i + 1 < SCHUNK)
            __builtin_prefetch(Kin + (rowbase + s + 1) * Dc + col4, 0, 0);

        float4 row = *reinterpret_cast<const float4*>(Kin + (rowbase + s) * Dc + col4);
        if (s == tstep)   // block-uniform branch; data override, pointers stay global
            row = *reinterpret_cast<const float4*>(kprow);

        *reinterpret_cast<float4*>(Kout + (rowbase + s) * Dc + col4) = row;

        float p = q0 * row.x + q1 * row.y + q2 * row.z + q3 * row.w;
        p += __shfl_xor(p, 8, 16);
        p += __shfl_xor(p, 4, 16);
        p += __shfl_xor(p, 2, 16);
        p += __shfl_xor(p, 1, 16);
        if (lane16 == 0)
            logits[((size_t)bp * Hc + head) * Sc + s] = p;
    }
}

// ---------------------------------------------------------------------------
// In-place softmax over S=1024 for each of the 2048 (bp,h) groups.
// block = 128 threads, 8 elements each (strided, coalesced).
// ---------------------------------------------------------------------------
__global__ __launch_bounds__(128)
void softmax_inplace(float* __restrict__ logits)
{
    const size_t base = (size_t)blockIdx.x * Sc;
    const int tid  = threadIdx.x;
    const int lane = tid & 31;
    const int wid  = tid >> 5;
    __shared__ float sred[4];

    float v[8];
#pragma unroll
    for (int j = 0; j < 8; ++j) v[j] = logits[base + tid + j * 128];

    float m = v[0];
#pragma unroll
    for (int j = 1; j < 8; ++j) m = fmaxf(m, v[j]);
#pragma unroll
    for (int off = 16; off >= 1; off >>= 1) m = fmaxf(m, __shfl_xor(m, off, 32));
    if (lane == 0) sred[wid] = m;
    __syncthreads();
    m = fmaxf(fmaxf(sred[0], sred[1]), fmaxf(sred[2], sred[3]));
    __syncthreads();

    float s = 0.f;
#pragma unroll
    for (int j = 0; j < 8; ++j) { v[j] = __expf(v[j] - m); s += v[j]; }
#pragma unroll
    for (int off = 16; off >= 1; off >>= 1) s += __shfl_xor(s, off, 32);
    if (lane == 0) sred[wid] = s;
    __syncthreads();
    s = sred[0] + sred[1] + sred[2] + sred[3];
    const float inv = 1.0f / s;

#pragma unroll
    for (int j = 0; j < 8; ++j) logits[base + tid + j * 128] = v[j] * inv;
}

// ---------------------------------------------------------------------------
// V pass: stream 128 rows of V per block; write V_out (row t replaced by vp
// via data select), accumulate zh partials per chunk.  Attn weights for the
// chunk staged in LDS.  grid = (NCHUNK, BP), block = 256.
// ---------------------------------------------------------------------------
__global__ __launch_bounds__(256)
void attn_v_pass(const float* __restrict__ Vin, const float* __restrict__ vp,
                 const float* __restrict__ attn, const int* __restrict__ tptr,
                 float* __restrict__ Vout, float* __restrict__ partials)
{
    const int bp    = blockIdx.y;
    const int chunk = blockIdx.x;
    const int s0    = chunk * SCHUNK;
    const int tid   = threadIdx.x;
    const int col4  = tid * 4;
    const int head  = tid >> 4;
    const int tstep = *tptr;

    __shared__ float satt[Hc * SCHUNK];  // 8 KB
    for (int idx = tid; idx < Hc * SCHUNK; idx += 256) {
        const int h = idx >> 7, i = idx & (SCHUNK - 1);
        satt[idx] = attn[((size_t)bp * Hc + h) * Sc + s0 + i];
    }
    __syncthreads();

    const float* vprow = vp + (size_t)bp * Dc + col4;
    const size_t rowbase = (size_t)bp * Sc;
    float a0 = 0.f, a1 = 0.f, a2 = 0.f, a3 = 0.f;

    for (int i = 0; i < SCHUNK; ++i) {
        const int s = s0 + i;
        if (i + 1 < SCHUNK)
            __builtin_prefetch(Vin + (rowbase + s + 1) * Dc + col4, 0, 0);

        float4 row = *reinterpret_cast<const float4*>(Vin + (rowbase + s) * Dc + col4);
        if (s == tstep)
            row = *reinterpret_cast<const float4*>(vprow);

        *reinterpret_cast<float4*>(Vout + (rowbase + s) * Dc + col4) = row;

        const float a = satt[head * SCHUNK + i];
        a0 += a * row.x; a1 += a * row.y; a2 += a * row.z; a3 += a * row.w;
    }

    float* dst = partials + ((size_t)bp * NCHUNK + chunk) * Dc + col4;
    dst[0] = a0; dst[1] = a1; dst[2] = a2; dst[3] = a3;
}

// ---------------------------------------------------------------------------
// Reduce the NCHUNK zh partials -> mean_z (BP x D).
// ---------------------------------------------------------------------------
__global__ __launch_bounds__(256)
void reduce_partials(const float* __restrict__ partials, float* __restrict__ meanz)
{
    const int idx = blockIdx.x * 256 + threadIdx.x;   // 0..131071
    const int bp = idx >> 10;
    const int d  = idx & (Dc - 1);
    float s = 0.f;
#pragma unroll
    for (int c = 0; c < NCHUNK; ++c)
        s += partials[((size_t)bp * NCHUNK + c) * Dc + d];
    meanz[(size_t)bp * Dc + d] = s;
}

// ---------------------------------------------------------------------------

extern "C" void kernel_launch(void* const* d_in, const int* in_sizes, int n_in,
                              void* d_out, int out_size, void* d_ws, size_t ws_size,
                              hipStream_t stream)
{
    const float* q  = (const float*)d_in[0];
    const float* k  = (const float*)d_in[1];
    const float* v  = (const float*)d_in[2];
    const float* Kc = (const float*)d_in[3];
    const float* Vc = (const float*)d_in[4];
    const float* Wq = (const float*)d_in[5];
    const float* bq = (const float*)d_in[6];
    const float* Wk = (const float*)d_in[7];
    const float* bk = (const float*)d_in[8];
    const float* Wv = (const float*)d_in[9];
    const float* bv = (const float*)d_in[10];
    const float* Wz = (const float*)d_in[11];
    const float* bz = (const float*)d_in[12];
    const float* nq = (const float*)d_in[13];
    const float* nk = (const float*)d_in[14];
    const float* nv = (const float*)d_in[15];
    const float* nz = (const float*)d_in[16];
    const int* tstep = (const int*)d_in[17];

    // Output layout (flat, return order): z | K_out | V_out | attn
    float* out      = (float*)d_out;
    float* z_out    = out;
    float* K_out    = out + (size_t)BP * Dc;
    float* V_out    = K_out + (size_t)BP * Sc * Dc;
    float* attn_out = V_out + (size_t)BP * Sc * Dc;

    // Workspace: qp | kp | vp | zh partials | mean_z  (~6 MB)
    float* ws  = (float*)d_ws;
    float* qp  = ws;
    float* kp  = qp + (size_t)BP * Dc;
    float* vp  = kp + (size_t)BP * Dc;
    float* pzh = vp + (size_t)BP * Dc;                 // BP * NCHUNK * D
    float* mz  = pzh + (size_t)BP * NCHUNK * Dc;

    const dim3 ggrid(Dc / 16, BP / 16);

    // 1) q/k/v projections (WMMA fp32), one launch per matrix
    proj_gemm_wmma<<<ggrid, 32, 0, stream>>>(q, Wq, bq, nq, qp);
    proj_gemm_wmma<<<ggrid, 32, 0, stream>>>(k, Wk, bk, nk, kp);
    proj_gemm_wmma<<<ggrid, 32, 0, stream>>>(v, Wv, bv, nv, vp);

    // 2) K copy + logits (fused single read of K)
    attn_k_pass<<<dim3(NCHUNK, BP), 256, 0, stream>>>(
        Kc, kp, qp, tstep, K_out, attn_out);

    // 3) softmax in place on the attn output region
    softmax_inplace<<<BP * Hc, 128, 0, stream>>>(attn_out);

    // 4) V copy + zh partials (fused single read of V)
    attn_v_pass<<<dim3(NCHUNK, BP), 256, 0, stream>>>(
        Vc, vp, attn_out, tstep, V_out, pzh);

    // 5) reduce partials -> mean_z
    reduce_partials<<<(BP * Dc) / 256, 256, 0, stream>>>(pzh, mz);

    // 6) z = mean_z @ Wz + bz + sigma*noise_z (WMMA fp32)
    proj_gemm_wmma<<<ggrid, 32, 0, stream>>>(mz, Wz, bz, nz, z_out);
}